// ShortConv_15109694947884
// MI455X (gfx1250) — compile-verified
//
#include <hip/hip_runtime.h>
#include <cstdint>

// Problem constants (match reference)
#define B_   4
#define S_   4096
#define M_   4
#define H_   1024
#define K_   4
#define T_   16          // output timesteps per block
#define HALO 3           // K-1 causal halo rows
#define ROWS (T_ + HALO) // 19 LDS rows
#define EPSF 1e-5f

typedef uint32_t v4u __attribute__((ext_vector_type(4)));
typedef int      v8i __attribute__((ext_vector_type(8)));
typedef int      v4i __attribute__((ext_vector_type(4)));

__launch_bounds__(256, 2)
__global__ void shortconv_tdm_kernel(const float* __restrict__ x,
                                     const float* __restrict__ norm_w,
                                     const float* __restrict__ conv_w,
                                     float* __restrict__ out)
{
    __shared__ float tile[ROWS][H_];   // 19 * 4KB = 76KB raw x rows
    __shared__ float rms_s[ROWS];      // per-row rsqrt(mean(x^2)+eps)

    const int tiles_per_bm = S_ / T_;          // 256
    const int tile_id = blockIdx.x % tiles_per_bm;
    const int bm      = blockIdx.x / tiles_per_bm;
    const int m = bm % M_;
    const int b = bm / M_;
    const int s0 = tile_id * T_;
    const int halo   = (s0 == 0) ? 0 : HALO;   // no rows before t=0
    const int gs     = s0 - halo;              // first global row fetched
    const int nrows  = T_ + halo;              // rows DMA'd
    const int ldsr0  = HALO - halo;            // LDS row where DMA lands

    const int tid = threadIdx.x;

    // ---------------- Tensor Data Mover descriptor (D#) ----------------
    // 2-D tile: dim0 = H contiguous floats, dim1 = nrows strided by M*H.
    const uint64_t ga = (uint64_t)(uintptr_t)(x + (((uint64_t)(b * S_ + gs) * M_ + m) * (uint64_t)H_));
    const uint32_t lds_addr = (uint32_t)(uintptr_t)(&tile[ldsr0][0]);

    v4u g0;
    g0.x = 1u;                                          // count=1, user mode
    g0.y = lds_addr;                                    // lds_addr [63:32]
    g0.z = (uint32_t)ga;                                // global_addr [95:64]
    g0.w = (uint32_t)((ga >> 32) & 0x01FFFFFFu)         // global_addr [120:96]
         | (2u << 30);                                  // type = 2 (image)

    const uint32_t td0    = H_;                         // tensor_dim0 (elements)
    const uint32_t td1    = (uint32_t)(S_ - gs);        // rows remaining (OOB clamp)
    const uint32_t tile0  = H_;                         // tile_dim0
    const uint32_t tile1  = (uint32_t)nrows;            // tile_dim1
    const uint64_t strd0  = (uint64_t)(M_ * H_);        // tensor_dim0_stride (elements)

    v8i g1;
    g1[0] = (int)(2u << 16);                                    // data_size=4B; mask=0
    g1[1] = (int)((td0 & 0xFFFFu) << 16);                       // abar=0 | td0.lo16
    g1[2] = (int)((td0 >> 16) | ((td1 & 0xFFFFu) << 16));       // td0.hi16 | td1.lo16
    g1[3] = (int)((td1 >> 16) | ((tile0 & 0xFFFFu) << 16));     // td1.hi16 | tile_dim0
    g1[4] = (int)(tile1 & 0xFFFFu);                             // tile_dim1 | tile_dim2=0
    g1[5] = (int)(uint32_t)strd0;                               // stride0.lo32
    g1[6] = (int)((uint32_t)(strd0 >> 32) & 0xFFFFu);           // stride0.hi16 | stride1.lo16=0
    g1[7] = 0;                                                  // stride1.hi32=0
    v4i g2 = {0, 0, 0, 0};
    v4i g3 = {0, 0, 0, 0};
    v8i g4 = {0, 0, 0, 0, 0, 0, 0, 0};  // extra descriptor group (unused, 2-D tile)

    if (tid < 32) {  // wave 0 issues the DMA once per block, waits TENSORcnt
        __builtin_amdgcn_tensor_load_to_lds(g0, g1, g2, g3, g4, 0);
        __builtin_amdgcn_s_wait_tensorcnt(0);
    }

    // Zero the causal pad rows at sequence start (disjoint from TDM target):
    // 3 rows x 256 float4 -> exactly one ds_store_b128 per thread per row.
    if (s0 == 0) {
        const float4 z = {0.0f, 0.0f, 0.0f, 0.0f};
        #pragma unroll
        for (int r = 0; r < HALO; ++r)
            ((float4*)&tile[r][0])[tid] = z;
    }
    __syncthreads();

    // ---------------- per-row RMS (one wave32 per row) ----------------
    const int wave = tid >> 5;
    const int lane = tid & 31;
    for (int r = wave; r < ROWS; r += 8) {
        const float4* row4 = (const float4*)&tile[r][0];
        float s = 0.0f;
        #pragma unroll
        for (int i = 0; i < H_ / (32 * 4); ++i) {   // 8 x ds_load_b128
            float4 v = row4[lane + i * 32];
            s += v.x * v.x + v.y * v.y + v.z * v.z + v.w * v.w;
        }
        #pragma unroll
        for (int off = 16; off > 0; off >>= 1)      // wave32 butterfly
            s += __shfl_xor(s, off);
        if (lane == 0)
            rms_s[r] = rsqrtf(s * (1.0f / (float)H_) + EPSF);
    }
    __syncthreads();

    // ---------------- conv + silu + residual ----------------
    const int h = tid * 4;                          // this thread's float4 of H
    const float4 nw = *(const float4*)&norm_w[m * H_ + h];

    // Fold norm weight into the conv taps: cwn[i].j = conv_w[h+i][j] * norm_w[h+i]
    float4 cwn[4];
    {
        const float nwe[4] = {nw.x, nw.y, nw.z, nw.w};
        #pragma unroll
        for (int i = 0; i < 4; ++i) {
            float4 c = *(const float4*)&conv_w[(size_t)(m * H_ + h + i) * K_];
            cwn[i].x = c.x * nwe[i];
            cwn[i].y = c.y * nwe[i];
            cwn[i].z = c.z * nwe[i];
            cwn[i].w = c.w * nwe[i];
        }
    }

    // Rolling window of rms-scaled rows: u_j = rms[row] * x[row]
    float4 u0, u1, u2;
    {
        float4 a0 = *(const float4*)&tile[0][h];
        float4 a1 = *(const float4*)&tile[1][h];
        float4 a2 = *(const float4*)&tile[2][h];
        float q0 = rms_s[0], q1 = rms_s[1], q2 = rms_s[2];
        u0.x = q0 * a0.x; u0.y = q0 * a0.y; u0.z = q0 * a0.z; u0.w = q0 * a0.w;
        u1.x = q1 * a1.x; u1.y = q1 * a1.y; u1.z = q1 * a1.z; u1.w = q1 * a1.w;
        u2.x = q2 * a2.x; u2.y = q2 * a2.y; u2.z = q2 * a2.z; u2.w = q2 * a2.w;
    }

    const uint64_t outbase = ((uint64_t)(b * S_ + s0) * M_ + m) * (uint64_t)H_ + h;

    #pragma unroll 4
    for (int t = 0; t < T_; ++t) {
        float4 r3 = *(const float4*)&tile[t + HALO][h];   // raw x (also residual)
        const float q3 = rms_s[t + HALO];
        float4 u3;
        u3.x = q3 * r3.x; u3.y = q3 * r3.y; u3.z = q3 * r3.z; u3.w = q3 * r3.w;

        float4 a;   // 16 FMAs
        a.x = fmaf(cwn[0].w, u3.x, fmaf(cwn[0].z, u2.x, fmaf(cwn[0].y, u1.x, cwn[0].x * u0.x)));
        a.y = fmaf(cwn[1].w, u3.y, fmaf(cwn[1].z, u2.y, fmaf(cwn[1].y, u1.y, cwn[1].x * u0.y)));
        a.z = fmaf(cwn[2].w, u3.z, fmaf(cwn[2].z, u2.z, fmaf(cwn[2].y, u1.z, cwn[2].x * u0.z)));
        a.w = fmaf(cwn[3].w, u3.w, fmaf(cwn[3].z, u2.w, fmaf(cwn[3].y, u1.w, cwn[3].x * u0.w)));

        float4 o;   // silu(a) + raw x residual
        o.x = a.x / (1.0f + __expf(-a.x)) + r3.x;
        o.y = a.y / (1.0f + __expf(-a.y)) + r3.y;
        o.z = a.z / (1.0f + __expf(-a.z)) + r3.z;
        o.w = a.w / (1.0f + __expf(-a.w)) + r3.w;

        *(float4*)&out[outbase + (uint64_t)t * (M_ * H_)] = o;

        u0 = u1; u1 = u2; u2 = u3;
    }
}

extern "C" void kernel_launch(void* const* d_in, const int* in_sizes, int n_in,
                              void* d_out, int out_size, void* d_ws, size_t ws_size,
                              hipStream_t stream) {
    const float* x      = (const float*)d_in[0];
    const float* norm_w = (const float*)d_in[1];
    const float* conv_w = (const float*)d_in[2];
    float* out = (float*)d_out;

    const int nblocks = B_ * M_ * (S_ / T_);   // 4096 blocks
    shortconv_tdm_kernel<<<nblocks, 256, 0, stream>>>(x, norm_w, conv_w, out);
}